// EMALayer_88716844466540
// MI455X (gfx1250) — compile-verified
//
#include <hip/hip_runtime.h>

// EMA scan via chunked linear-recurrence closed form on CDNA5 (gfx1250).
//
//   chunk (T=16) local:  y_t = sum_{s<=t} K_d(t-s) x_s + sum_n d_n a_n^{t+1} h0_n
//   K_d(tau) = sum_n d_n (1-a_n) a_n^tau
//   chunk-end (zero-init): G[n] = sum_s a_n^{15-s}(1-a_n) x_s
//   inter-chunk scan:      h0[j+1] = a^16 * h0[j] + G[j]
//
// Block = 512 threads = 16 waves; wave w owns channel d = dbase + w.
// Block covers (b, 16 channels) and walks 16 superchunks of 256 timesteps,
// carrying state in registers.  Per superchunk per wave: 4x v_wmma_f32_16x16x4_f32
// (G = W*X), a 16-step scalar scan, 8x wmma (Y = [M|P] * [X;H0]).
//
// Data movement: double-buffered GLOBAL_LOAD_ASYNC_TO_LDS_B32 (per-lane scatter,
// allows the transposed [d][t] LDS layout) and GLOBAL_STORE_ASYNC_FROM_LDS_B32,
// ASYNCcnt-tracked and overlapped with WMMA work; __has_builtin-guarded with a
// synchronous fallback.

#define D_MODEL 1024
#define D_STATE 16
#define SEQ_LEN 4096
#define BATCH   8
#define T_CHUNK 16                    // WMMA M dim (timesteps per chunk)
#define N_CHUNK 16                    // WMMA N dim (chunks per superchunk)
#define SC_LEN  (T_CHUNK * N_CHUNK)   // 256 timesteps per superchunk
#define N_SC    (SEQ_LEN / SC_LEN)    // 16 superchunks
#define D_TILE  16                    // channels per block (one per wave)
#define THREADS (32 * D_TILE)         // 512

typedef float v2f __attribute__((ext_vector_type(2)));
typedef float v8f __attribute__((ext_vector_type(8)));

#if __has_builtin(__builtin_amdgcn_global_load_async_to_lds_b32) && \
    __has_builtin(__builtin_amdgcn_global_store_async_from_lds_b32)
#define USE_ASYNC 1
#else
#define USE_ASYNC 0
#endif

#if __has_builtin(__builtin_amdgcn_s_wait_asynccnt)
#define WAIT_ASYNC() __builtin_amdgcn_s_wait_asynccnt(0)
#else
#define WAIT_ASYNC() asm volatile("s_wait_asynccnt 0" ::: "memory")
#endif

#define GAS __attribute__((address_space(1)))   // global ("__device__")
#define LAS __attribute__((address_space(3)))   // LDS    ("__shared__")

// ---- LDS pool layout (floats) ----
#define XS_STRIDE 260                             // 256 t + pad, per channel
#define XS_BUF    (D_TILE * XS_STRIDE)            // 4160: one x tile [16 d][260]
#define XS_OFF    0                               // two buffers: 2*4160
#define ADS_OFF   (XS_OFF + 2 * XS_BUF)           // [D_TILE][16] sigmoid(alpha)
#define DDS_OFF   (ADS_OFF + D_TILE * 16)         // [D_TILE][16] sigmoid(delta)
#define KTAB_OFF  (DDS_OFF + D_TILE * 16)         // [D_TILE][16] K_d(tau)
#define U1_OFF    (KTAB_OFF + D_TILE * 16)        // apow[16][16][17] -> h0s[16][16][17]
#define U2_OFF    (U1_OFF + D_TILE * 272)         // gs[16][16][17]   -> ys[256][17]
#define POOL_SZ   (U2_OFF + SC_LEN * 17)          // 17792 floats = 71.2 KB

__global__ __launch_bounds__(THREADS)
void ema_scan_wmma(const float* __restrict__ x, const float* __restrict__ alpha,
                   const float* __restrict__ delta, const float* __restrict__ gamma,
                   const float* __restrict__ beta, float* __restrict__ y)
{
    __shared__ float lds[POOL_SZ];

    const int tid  = threadIdx.x;
    const int wv   = tid >> 5;          // wave id = local channel
    const int lane = tid & 31;
    const int m    = lane & 15;         // WMMA row within half (t for Ay, n for Ag)
    const bool hi  = lane >= 16;
    const int dbase = blockIdx.x * D_TILE;
    const int b     = blockIdx.y;
    const int d     = dbase + wv;

    // ---------------- per-block preamble: tables ----------------
    if (lane < 16) {
        float av = 1.0f / (1.0f + __expf(-alpha[d * D_STATE + lane]));
        float dv = 1.0f / (1.0f + __expf(-delta[d * D_STATE + lane]));
        lds[ADS_OFF + wv * 16 + lane] = av;
        lds[DDS_OFF + wv * 16 + lane] = dv;
        float p = 1.0f;
        #pragma unroll
        for (int e = 0; e <= 16; ++e) {             // apow[n][e] = a_n^e
            lds[U1_OFF + wv * 272 + lane * 17 + e] = p;
            p *= av;
        }
    }
    __syncthreads();
    if (lane < 16) {                                 // K_d(tau), tau = lane
        float s = 0.0f;
        #pragma unroll
        for (int n = 0; n < 16; ++n) {
            s += lds[DDS_OFF + wv * 16 + n] * (1.0f - lds[ADS_OFF + wv * 16 + n]) *
                 lds[U1_OFF + wv * 272 + n * 17 + lane];
        }
        lds[KTAB_OFF + wv * 16 + lane] = s;
    }
    __syncthreads();

    // A matrices in registers (superchunk-invariant).
    // fp32 A 16x4 layout: lanes0-15: vgpr0=K0, vgpr1=K1; lanes16-31: K2, K3.
    v2f Ay[8];   // [ M_d (t,s) | P_d (t,n) ], 16x32, row m = t
    v2f Ag[4];   // W_d (n,s), 16x16, row m = n
    #pragma unroll
    for (int kk = 0; kk < 8; ++kk) {
        float v0, v1;
        #pragma unroll
        for (int c = 0; c < 2; ++c) {
            int k = 4 * kk + (hi ? 2 : 0) + c;
            float val;
            if (k < 16) {                      // causal Toeplitz conv matrix
                val = (m >= k) ? lds[KTAB_OFF + wv * 16 + (m - k)] : 0.0f;
            } else {                           // state-carry matrix d_n * a_n^{t+1}
                int n = k - 16;
                val = lds[DDS_OFF + wv * 16 + n] * lds[U1_OFF + wv * 272 + n * 17 + (m + 1)];
            }
            if (c == 0) v0 = val; else v1 = val;
        }
        Ay[kk].x = v0; Ay[kk].y = v1;
    }
    #pragma unroll
    for (int kk = 0; kk < 4; ++kk) {
        float v0, v1;
        #pragma unroll
        for (int c = 0; c < 2; ++c) {
            int k = 4 * kk + (hi ? 2 : 0) + c;  // s index
            float val = lds[U1_OFF + wv * 272 + m * 17 + (15 - k)] *
                        (1.0f - lds[ADS_OFF + wv * 16 + m]);
            if (c == 0) v0 = val; else v1 = val;
        }
        Ag[kk].x = v0; Ag[kk].y = v1;
    }
    const float a16 = lds[U1_OFF + wv * 272 + m * 17 + 16];   // a_n^16 (scan lanes)
    const float gam = gamma[d];
    const float bet = beta[d];
    float hcarry = 0.0f;                                       // state, lanes 0-15 (n)
    __syncthreads();   // after this, U1 region is reused as h0s

    const float* xb = x + (size_t)b * SEQ_LEN * D_MODEL + dbase;
    float*       yb = y + (size_t)b * SEQ_LEN * D_MODEL + dbase;

    int buf = 0;

#if USE_ASYNC
    // Preload tile 0: per-lane b32 scatter global[row][dl] -> lds[dl][row].
    {
        #pragma unroll
        for (int i = 0; i < 8; ++i) {
            int idx = tid + i * THREADS;
            int row = idx >> 4, dl = idx & 15;
            __builtin_amdgcn_global_load_async_to_lds_b32(
                (GAS int*)(xb + (size_t)row * D_MODEL + dl),
                (LAS int*)(&lds[XS_OFF + dl * XS_STRIDE + row]), 0, 0);
        }
    }
#endif

    // ---------------- sequential superchunk loop ----------------
    for (int sc = 0; sc < N_SC; ++sc) {
        float* xs = &lds[XS_OFF + buf * XS_BUF];

#if USE_ASYNC
        WAIT_ASYNC();          // own tile loads + own previous y stores done
        __syncthreads();       // => ALL waves' async traffic has landed
        if (sc + 1 < N_SC) {   // kick off next tile into the other buffer
            const float* xn = xb + (size_t)(sc + 1) * SC_LEN * D_MODEL;
            float* xd = &lds[XS_OFF + (buf ^ 1) * XS_BUF];
            #pragma unroll
            for (int i = 0; i < 8; ++i) {
                int idx = tid + i * THREADS;
                int row = idx >> 4, dl = idx & 15;
                __builtin_amdgcn_global_load_async_to_lds_b32(
                    (GAS int*)(xn + (size_t)row * D_MODEL + dl),
                    (LAS int*)(&xd[dl * XS_STRIDE + row]), 0, 0);
            }
        }
#else
        {
            const float* xt = xb + (size_t)sc * SC_LEN * D_MODEL;
            #pragma unroll
            for (int i = 0; i < 8; ++i) {
                int idx = tid + i * THREADS;
                int row = idx >> 4, dl = idx & 15;
                xs[dl * XS_STRIDE + row] =
                    __builtin_nontemporal_load(xt + (size_t)row * D_MODEL + dl);
            }
            if (sc + 1 < N_SC && tid < SC_LEN) {
                __builtin_prefetch(xt + (size_t)SC_LEN * D_MODEL + (size_t)tid * D_MODEL, 0, 1);
            }
        }
        __syncthreads();
#endif

        const int j = m;   // WMMA column = chunk index

        // G(n, chunk) = W_d * X    (4 x wmma_f32_16x16x4_f32, K=16)
        v8f g = {};
        #pragma unroll
        for (int kk = 0; kk < 4; ++kk) {
            int k0 = 4 * kk + (hi ? 2 : 0);
            v2f bx;
            bx.x = xs[wv * XS_STRIDE + j * 16 + k0];
            bx.y = xs[wv * XS_STRIDE + j * 16 + k0 + 1];
            g = __builtin_amdgcn_wmma_f32_16x16x4_f32(false, Ag[kk], false, bx,
                                                      (short)0, g, false, false);
        }
        // spill G tile: gs[n][chunk]
        #pragma unroll
        for (int r = 0; r < 8; ++r) {
            int n = r + (hi ? 8 : 0);
            lds[U2_OFF + wv * 272 + n * 17 + j] = g[r];
        }
        __syncthreads();

        // inter-chunk scan (lane n): h0[j] stored, h = a^16 h + G[j]
        if (lane < 16) {
            float h = hcarry;
            #pragma unroll
            for (int jj = 0; jj < 16; ++jj) {
                lds[U1_OFF + wv * 272 + m * 17 + jj] = h;          // h0s[n][jj]
                h = a16 * h + lds[U2_OFF + wv * 272 + m * 17 + jj];
            }
            hcarry = h;
        }
        __syncthreads();

        // Y(t, chunk) = [M|P] * [X; H0]   (8 x wmma, K=32)
        v8f acc = {};
        #pragma unroll
        for (int kk = 0; kk < 8; ++kk) {
            int k0 = 4 * kk + (hi ? 2 : 0);
            v2f bx;
            bx.x = (k0 < 16) ? xs[wv * XS_STRIDE + j * 16 + k0]
                             : lds[U1_OFF + wv * 272 + (k0 - 16) * 17 + j];
            int k1 = k0 + 1;
            bx.y = (k1 < 16) ? xs[wv * XS_STRIDE + j * 16 + k1]
                             : lds[U1_OFF + wv * 272 + (k1 - 16) * 17 + j];
            acc = __builtin_amdgcn_wmma_f32_16x16x4_f32(false, Ay[kk], false, bx,
                                                        (short)0, acc, false, false);
        }

        // gamma/beta + transpose through LDS: ys[t_local_256][d]
        #pragma unroll
        for (int r = 0; r < 8; ++r) {
            int tl  = r + (hi ? 8 : 0);
            int row = j * 16 + tl;
            lds[U2_OFF + row * 17 + wv] = acc[r] * gam + bet;
        }
        __syncthreads();

        float* yt = yb + (size_t)sc * SC_LEN * D_MODEL;
#if USE_ASYNC
        // async LDS -> global y stores; drained by next iteration's WAIT_ASYNC
        // (+ barrier) before anyone rewrites U2, or by S_ENDPGM's implicit idle.
        #pragma unroll
        for (int i = 0; i < 8; ++i) {
            int idx = tid + i * THREADS;
            int row = idx >> 4, dl = idx & 15;
            __builtin_amdgcn_global_store_async_from_lds_b32(
                (GAS int*)(yt + (size_t)row * D_MODEL + dl),
                (LAS int*)(&lds[U2_OFF + row * 17 + dl]), 0, 0);
        }
#else
        #pragma unroll
        for (int i = 0; i < 8; ++i) {
            int idx = tid + i * THREADS;
            int row = idx >> 4, dl = idx & 15;
            __builtin_nontemporal_store(lds[U2_OFF + row * 17 + dl],
                                        yt + (size_t)row * D_MODEL + dl);
        }
        __syncthreads();
#endif
        buf ^= 1;
    }
}

extern "C" void kernel_launch(void* const* d_in, const int* in_sizes, int n_in,
                              void* d_out, int out_size, void* d_ws, size_t ws_size,
                              hipStream_t stream) {
    (void)in_sizes; (void)n_in; (void)out_size; (void)d_ws; (void)ws_size;
    const float* x     = (const float*)d_in[0];
    const float* alpha = (const float*)d_in[1];
    const float* delta = (const float*)d_in[2];
    const float* gamma = (const float*)d_in[3];
    const float* beta  = (const float*)d_in[4];
    float* yout = (float*)d_out;

    dim3 grid(D_MODEL / D_TILE, BATCH);   // 64 x 8 blocks, 16 waves each
    ema_scan_wmma<<<grid, THREADS, 0, stream>>>(x, alpha, delta, gamma, beta, yout);
}